// MeshGraphNetBlock_37915971289748
// MI455X (gfx1250) — compile-verified
//
#include <hip/hip_runtime.h>
#include <hip/hip_bf16.h>

typedef float v2f __attribute__((ext_vector_type(2)));
typedef float v8f __attribute__((ext_vector_type(8)));

#define WIDTH 128
#define LDSP  132   // padded LDS row stride (floats)

__device__ __forceinline__ v8f wmma_k4(v2f a, v2f b, v8f c) {
    return __builtin_amdgcn_wmma_f32_16x16x4_f32(false, a, false, b, (short)0, c,
                                                 false, false);
}

__device__ __forceinline__ float gelu_exact(float v) {
    return 0.5f * v * (1.0f + erff(v * 0.70710678118654752f));
}

__device__ __forceinline__ void ds_phase_fence() {
    asm volatile("s_wait_dscnt 0" ::: "memory");
}

// ---------------------------------------------------------------------------
__global__ void zero_ws_kernel(float* __restrict__ p, long long n) {
    long long i = (long long)blockIdx.x * blockDim.x + threadIdx.x;
    long long stride = (long long)gridDim.x * blockDim.x;
    for (; i < n; i += stride) p[i] = 0.0f;
}

// W: K x C row-major  ->  Wt: C x K row-major
__global__ void transpose_kernel(const float* __restrict__ W,
                                 float* __restrict__ Wt, int K, int C) {
    int total = K * C;
    for (int idx = blockIdx.x * blockDim.x + threadIdx.x; idx < total;
         idx += gridDim.x * blockDim.x) {
        int k = idx / C, c = idx - k * C;
        Wt[c * K + k] = W[k * C + c];
    }
}

// ---------------------------------------------------------------------------
// Edge kernel: one wave32 per 32-edge tile (2 M-tiles x 2 N-tiles blocking).
// ---------------------------------------------------------------------------
__global__ __launch_bounds__(32) void edge_kernel(
    const float* __restrict__ x, const float* __restrict__ ea,
    const int* __restrict__ eidx,
    const float* __restrict__ W1t, const float* __restrict__ b1,
    const float* __restrict__ W2t, const float* __restrict__ b2,
    const float* __restrict__ W3t, const float* __restrict__ b3,
    const float* __restrict__ gamma, const float* __restrict__ beta,
    float* __restrict__ out_edge, float* __restrict__ agg,
    float* __restrict__ degree, int E)
{
    __shared__ float hA[32 * LDSP];
    __shared__ float hB[32 * LDSP];
    __shared__ float s_mu[32], s_rs[32];
    __shared__ int   s_dst[32];

    const int lane = threadIdx.x;
    const int m    = lane & 15;
    const int half = lane >> 4;
    const int col  = lane & 15;
    const int k2   = 2 * half;

    const long long e0 = (long long)blockIdx.x * 32;
    long long eA = e0 + m;       if (eA >= E) eA = (long long)E - 1;
    long long eB = e0 + 16 + m;  if (eB >= E) eB = (long long)E - 1;

    const int sA = eidx[eA], dA = eidx[(long long)E + eA];
    const int sB = eidx[eB], dB = eidx[(long long)E + eB];

    const float* a0seg[3] = { x + (long long)sA * WIDTH,
                              x + (long long)dA * WIDTH,
                              ea + eA * WIDTH };
    const float* a1seg[3] = { x + (long long)sB * WIDTH,
                              x + (long long)dB * WIDTH,
                              ea + eB * WIDTH };
    s_dst[lane] = (lane < 16) ? dA : dB;

    // ---- GEMM1: (32x384) @ (384x128) + GELU -> hA -------------------------
    for (int nt = 0; nt < 4; ++nt) {
        const int cn0 = nt * 32 + col, cn1 = cn0 + 16;
        v8f a00, a01, a10, a11;
        {
            const float bb0 = b1[cn0], bb1 = b1[cn1];
            for (int i = 0; i < 8; ++i) { a00[i] = bb0; a01[i] = bb1;
                                          a10[i] = bb0; a11[i] = bb1; }
        }
#pragma unroll
        for (int seg = 0; seg < 3; ++seg) {
            const float* pa0 = a0seg[seg] + k2;
            const float* pa1 = a1seg[seg] + k2;
            const float* pb0 = W1t + cn0 * 384 + seg * 128 + k2;
            const float* pb1 = W1t + cn1 * 384 + seg * 128 + k2;
#pragma unroll 8
            for (int k = 0; k < 128; k += 4) {
                v2f fa0 = *(const v2f*)(pa0 + k);
                v2f fa1 = *(const v2f*)(pa1 + k);
                v2f fb0 = *(const v2f*)(pb0 + k);
                v2f fb1 = *(const v2f*)(pb1 + k);
                a00 = wmma_k4(fa0, fb0, a00);
                a01 = wmma_k4(fa0, fb1, a01);
                a10 = wmma_k4(fa1, fb0, a10);
                a11 = wmma_k4(fa1, fb1, a11);
            }
        }
        for (int i = 0; i < 8; ++i) {
            const int r0 = i + 8 * half, r1 = r0 + 16;
            hA[r0 * LDSP + cn0] = gelu_exact(a00[i]);
            hA[r0 * LDSP + cn1] = gelu_exact(a01[i]);
            hA[r1 * LDSP + cn0] = gelu_exact(a10[i]);
            hA[r1 * LDSP + cn1] = gelu_exact(a11[i]);
        }
    }
    ds_phase_fence();

    // ---- GEMM2: (32x128) @ (128x128) + GELU -> hB -------------------------
    for (int nt = 0; nt < 4; ++nt) {
        const int cn0 = nt * 32 + col, cn1 = cn0 + 16;
        v8f a00, a01, a10, a11;
        {
            const float bb0 = b2[cn0], bb1 = b2[cn1];
            for (int i = 0; i < 8; ++i) { a00[i] = bb0; a01[i] = bb1;
                                          a10[i] = bb0; a11[i] = bb1; }
        }
        const float* pa0 = &hA[m * LDSP + k2];
        const float* pa1 = &hA[(16 + m) * LDSP + k2];
        const float* pb0 = W2t + cn0 * 128 + k2;
        const float* pb1 = W2t + cn1 * 128 + k2;
#pragma unroll 8
        for (int k = 0; k < 128; k += 4) {
            v2f fa0 = *(const v2f*)(pa0 + k);
            v2f fa1 = *(const v2f*)(pa1 + k);
            v2f fb0 = *(const v2f*)(pb0 + k);
            v2f fb1 = *(const v2f*)(pb1 + k);
            a00 = wmma_k4(fa0, fb0, a00);
            a01 = wmma_k4(fa0, fb1, a01);
            a10 = wmma_k4(fa1, fb0, a10);
            a11 = wmma_k4(fa1, fb1, a11);
        }
        for (int i = 0; i < 8; ++i) {
            const int r0 = i + 8 * half, r1 = r0 + 16;
            hB[r0 * LDSP + cn0] = gelu_exact(a00[i]);
            hB[r0 * LDSP + cn1] = gelu_exact(a01[i]);
            hB[r1 * LDSP + cn0] = gelu_exact(a10[i]);
            hB[r1 * LDSP + cn1] = gelu_exact(a11[i]);
        }
    }
    ds_phase_fence();

    // ---- GEMM3: (32x128) @ (128x128) -> hA --------------------------------
    for (int nt = 0; nt < 4; ++nt) {
        const int cn0 = nt * 32 + col, cn1 = cn0 + 16;
        v8f a00, a01, a10, a11;
        {
            const float bb0 = b3[cn0], bb1 = b3[cn1];
            for (int i = 0; i < 8; ++i) { a00[i] = bb0; a01[i] = bb1;
                                          a10[i] = bb0; a11[i] = bb1; }
        }
        const float* pa0 = &hB[m * LDSP + k2];
        const float* pa1 = &hB[(16 + m) * LDSP + k2];
        const float* pb0 = W3t + cn0 * 128 + k2;
        const float* pb1 = W3t + cn1 * 128 + k2;
#pragma unroll 8
        for (int k = 0; k < 128; k += 4) {
            v2f fa0 = *(const v2f*)(pa0 + k);
            v2f fa1 = *(const v2f*)(pa1 + k);
            v2f fb0 = *(const v2f*)(pb0 + k);
            v2f fb1 = *(const v2f*)(pb1 + k);
            a00 = wmma_k4(fa0, fb0, a00);
            a01 = wmma_k4(fa0, fb1, a01);
            a10 = wmma_k4(fa1, fb0, a10);
            a11 = wmma_k4(fa1, fb1, a11);
        }
        for (int i = 0; i < 8; ++i) {
            const int r0 = i + 8 * half, r1 = r0 + 16;
            hA[r0 * LDSP + cn0] = a00[i];
            hA[r0 * LDSP + cn1] = a01[i];
            hA[r1 * LDSP + cn0] = a10[i];
            hA[r1 * LDSP + cn1] = a11[i];
        }
    }
    ds_phase_fence();

    // ---- LayerNorm stats: one lane per row (32 rows) ----------------------
    {
        float s = 0.0f, s2 = 0.0f;
        const float* hr = &hA[lane * LDSP];
        for (int c = 0; c < WIDTH; ++c) { float v = hr[c]; s += v; s2 += v * v; }
        const float mu  = s * (1.0f / WIDTH);
        const float var = s2 * (1.0f / WIDTH) - mu * mu;
        s_mu[lane] = mu;
        s_rs[lane] = rsqrtf(var + 1e-5f);
    }
    ds_phase_fence();

    // ---- normalize, write edge_update, scatter-add ------------------------
    for (int r = 0; r < 32; ++r) {
        const long long er = e0 + r;
        if (er >= E) break;
        const float mu = s_mu[r], rsg = s_rs[r];
        float* aggrow = agg + (long long)s_dst[r] * WIDTH;
        for (int c = lane; c < WIDTH; c += 32) {
            float v = (hA[r * LDSP + c] - mu) * rsg * gamma[c] + beta[c];
            out_edge[er * WIDTH + c] = v;
            unsafeAtomicAdd(&aggrow[c], v);
        }
    }
    if ((e0 + lane) < E)
        unsafeAtomicAdd(&degree[s_dst[lane]], 1.0f);
}

// ---------------------------------------------------------------------------
// Node kernel: one wave32 per 32-node tile (2M x 2N blocking).
// ---------------------------------------------------------------------------
__global__ __launch_bounds__(32) void node_kernel(
    const float* __restrict__ x, const float* __restrict__ agg,
    const float* __restrict__ degree,
    const float* __restrict__ W1t, const float* __restrict__ b1,
    const float* __restrict__ W2t, const float* __restrict__ b2,
    const float* __restrict__ W3t, const float* __restrict__ b3,
    const float* __restrict__ gamma, const float* __restrict__ beta,
    float* __restrict__ out_node, int N)
{
    __shared__ float hA[32 * LDSP];
    __shared__ float hB[32 * LDSP];
    __shared__ float s_mu[32], s_rs[32];

    const int lane = threadIdx.x;
    const int m    = lane & 15;
    const int half = lane >> 4;
    const int col  = lane & 15;
    const int k2   = 2 * half;

    const long long n0 = (long long)blockIdx.x * 32;
    long long nA = n0 + m;       if (nA >= N) nA = (long long)N - 1;
    long long nB = n0 + 16 + m;  if (nB >= N) nB = (long long)N - 1;

    const float* a0seg[2] = { x + nA * WIDTH, agg + nA * WIDTH };
    const float* a1seg[2] = { x + nB * WIDTH, agg + nB * WIDTH };
    const float rdeg0 = 1.0f / fmaxf(degree[nA], 1.0f);
    const float rdeg1 = 1.0f / fmaxf(degree[nB], 1.0f);

    // ---- GEMM1: (32x256) @ (256x128) + GELU -> hA -------------------------
    for (int nt = 0; nt < 4; ++nt) {
        const int cn0 = nt * 32 + col, cn1 = cn0 + 16;
        v8f a00, a01, a10, a11;
        {
            const float bb0 = b1[cn0], bb1 = b1[cn1];
            for (int i = 0; i < 8; ++i) { a00[i] = bb0; a01[i] = bb1;
                                          a10[i] = bb0; a11[i] = bb1; }
        }
#pragma unroll
        for (int seg = 0; seg < 2; ++seg) {
            const float* pa0 = a0seg[seg] + k2;
            const float* pa1 = a1seg[seg] + k2;
            const float* pb0 = W1t + cn0 * 256 + seg * 128 + k2;
            const float* pb1 = W1t + cn1 * 256 + seg * 128 + k2;
#pragma unroll 8
            for (int k = 0; k < 128; k += 4) {
                v2f fa0 = *(const v2f*)(pa0 + k);
                v2f fa1 = *(const v2f*)(pa1 + k);
                if (seg == 1) { fa0.x *= rdeg0; fa0.y *= rdeg0;
                                fa1.x *= rdeg1; fa1.y *= rdeg1; }
                v2f fb0 = *(const v2f*)(pb0 + k);
                v2f fb1 = *(const v2f*)(pb1 + k);
                a00 = wmma_k4(fa0, fb0, a00);
                a01 = wmma_k4(fa0, fb1, a01);
                a10 = wmma_k4(fa1, fb0, a10);
                a11 = wmma_k4(fa1, fb1, a11);
            }
        }
        for (int i = 0; i < 8; ++i) {
            const int r0 = i + 8 * half, r1 = r0 + 16;
            hA[r0 * LDSP + cn0] = gelu_exact(a00[i]);
            hA[r0 * LDSP + cn1] = gelu_exact(a01[i]);
            hA[r1 * LDSP + cn0] = gelu_exact(a10[i]);
            hA[r1 * LDSP + cn1] = gelu_exact(a11[i]);
        }
    }
    ds_phase_fence();

    // ---- GEMM2 -> hB ------------------------------------------------------
    for (int nt = 0; nt < 4; ++nt) {
        const int cn0 = nt * 32 + col, cn1 = cn0 + 16;
        v8f a00, a01, a10, a11;
        {
            const float bb0 = b2[cn0], bb1 = b2[cn1];
            for (int i = 0; i < 8; ++i) { a00[i] = bb0; a01[i] = bb1;
                                          a10[i] = bb0; a11[i] = bb1; }
        }
        const float* pa0 = &hA[m * LDSP + k2];
        const float* pa1 = &hA[(16 + m) * LDSP + k2];
        const float* pb0 = W2t + cn0 * 128 + k2;
        const float* pb1 = W2t + cn1 * 128 + k2;
#pragma unroll 8
        for (int k = 0; k < 128; k += 4) {
            v2f fa0 = *(const v2f*)(pa0 + k);
            v2f fa1 = *(const v2f*)(pa1 + k);
            v2f fb0 = *(const v2f*)(pb0 + k);
            v2f fb1 = *(const v2f*)(pb1 + k);
            a00 = wmma_k4(fa0, fb0, a00);
            a01 = wmma_k4(fa0, fb1, a01);
            a10 = wmma_k4(fa1, fb0, a10);
            a11 = wmma_k4(fa1, fb1, a11);
        }
        for (int i = 0; i < 8; ++i) {
            const int r0 = i + 8 * half, r1 = r0 + 16;
            hB[r0 * LDSP + cn0] = gelu_exact(a00[i]);
            hB[r0 * LDSP + cn1] = gelu_exact(a01[i]);
            hB[r1 * LDSP + cn0] = gelu_exact(a10[i]);
            hB[r1 * LDSP + cn1] = gelu_exact(a11[i]);
        }
    }
    ds_phase_fence();

    // ---- GEMM3 -> hA ------------------------------------------------------
    for (int nt = 0; nt < 4; ++nt) {
        const int cn0 = nt * 32 + col, cn1 = cn0 + 16;
        v8f a00, a01, a10, a11;
        {
            const float bb0 = b3[cn0], bb1 = b3[cn1];
            for (int i = 0; i < 8; ++i) { a00[i] = bb0; a01[i] = bb1;
                                          a10[i] = bb0; a11[i] = bb1; }
        }
        const float* pa0 = &hB[m * LDSP + k2];
        const float* pa1 = &hB[(16 + m) * LDSP + k2];
        const float* pb0 = W3t + cn0 * 128 + k2;
        const float* pb1 = W3t + cn1 * 128 + k2;
#pragma unroll 8
        for (int k = 0; k < 128; k += 4) {
            v2f fa0 = *(const v2f*)(pa0 + k);
            v2f fa1 = *(const v2f*)(pa1 + k);
            v2f fb0 = *(const v2f*)(pb0 + k);
            v2f fb1 = *(const v2f*)(pb1 + k);
            a00 = wmma_k4(fa0, fb0, a00);
            a01 = wmma_k4(fa0, fb1, a01);
            a10 = wmma_k4(fa1, fb0, a10);
            a11 = wmma_k4(fa1, fb1, a11);
        }
        for (int i = 0; i < 8; ++i) {
            const int r0 = i + 8 * half, r1 = r0 + 16;
            hA[r0 * LDSP + cn0] = a00[i];
            hA[r0 * LDSP + cn1] = a01[i];
            hA[r1 * LDSP + cn0] = a10[i];
            hA[r1 * LDSP + cn1] = a11[i];
        }
    }
    ds_phase_fence();

    // ---- LayerNorm stats --------------------------------------------------
    {
        float s = 0.0f, s2 = 0.0f;
        const float* hr = &hA[lane * LDSP];
        for (int c = 0; c < WIDTH; ++c) { float v = hr[c]; s += v; s2 += v * v; }
        const float mu  = s * (1.0f / WIDTH);
        const float var = s2 * (1.0f / WIDTH) - mu * mu;
        s_mu[lane] = mu;
        s_rs[lane] = rsqrtf(var + 1e-5f);
    }
    ds_phase_fence();

    // ---- normalize + residual write --------------------------------------
    for (int r = 0; r < 32; ++r) {
        const long long nr = n0 + r;
        if (nr >= N) break;
        const float mu = s_mu[r], rsg = s_rs[r];
        for (int c = lane; c < WIDTH; c += 32) {
            float v = (hA[r * LDSP + c] - mu) * rsg * gamma[c] + beta[c];
            out_node[nr * WIDTH + c] = x[nr * WIDTH + c] + v;
        }
    }
}

// ---------------------------------------------------------------------------
extern "C" void kernel_launch(void* const* d_in, const int* in_sizes, int n_in,
                              void* d_out, int out_size, void* d_ws, size_t ws_size,
                              hipStream_t stream) {
    const float* x    = (const float*)d_in[0];
    const float* ea   = (const float*)d_in[1];
    const int*   eidx = (const int*)d_in[2];
    const float* eW1  = (const float*)d_in[3];
    const float* eb1  = (const float*)d_in[4];
    const float* eW2  = (const float*)d_in[5];
    const float* eb2  = (const float*)d_in[6];
    const float* eW3  = (const float*)d_in[7];
    const float* eb3  = (const float*)d_in[8];
    const float* nW1  = (const float*)d_in[9];
    const float* nb1  = (const float*)d_in[10];
    const float* nW2  = (const float*)d_in[11];
    const float* nb2  = (const float*)d_in[12];
    const float* nW3  = (const float*)d_in[13];
    const float* nb3  = (const float*)d_in[14];
    const float* eg   = (const float*)d_in[15];
    const float* ebt  = (const float*)d_in[16];
    const float* ng   = (const float*)d_in[17];
    const float* nbt  = (const float*)d_in[18];

    const int N = in_sizes[0] / WIDTH;
    const int E = in_sizes[1] / WIDTH;

    float* out_node = (float*)d_out;
    float* out_edge = out_node + (long long)N * WIDTH;

    // workspace layout
    float* agg    = (float*)d_ws;                         // N*128
    float* degree = agg + (long long)N * WIDTH;           // N
    float* eW1t   = degree + N;                           // 128 x 384
    float* eW2t   = eW1t + 384 * 128;                     // 128 x 128
    float* eW3t   = eW2t + 128 * 128;
    float* nW1t   = eW3t + 128 * 128;                     // 128 x 256
    float* nW2t   = nW1t + 256 * 128;
    float* nW3t   = nW2t + 128 * 128;

    const long long zn = (long long)N * WIDTH + N;
    zero_ws_kernel<<<1024, 256, 0, stream>>>(agg, zn);

    transpose_kernel<<<192, 256, 0, stream>>>(eW1, eW1t, 384, 128);
    transpose_kernel<<<64,  256, 0, stream>>>(eW2, eW2t, 128, 128);
    transpose_kernel<<<64,  256, 0, stream>>>(eW3, eW3t, 128, 128);
    transpose_kernel<<<128, 256, 0, stream>>>(nW1, nW1t, 256, 128);
    transpose_kernel<<<64,  256, 0, stream>>>(nW2, nW2t, 128, 128);
    transpose_kernel<<<64,  256, 0, stream>>>(nW3, nW3t, 128, 128);

    const int etiles = (E + 31) / 32;
    edge_kernel<<<etiles, 32, 0, stream>>>(
        x, ea, eidx, eW1t, eb1, eW2t, eb2, eW3t, eb3, eg, ebt,
        out_edge, agg, degree, E);

    const int ntiles = (N + 31) / 32;
    node_kernel<<<ntiles, 32, 0, stream>>>(
        x, agg, degree, nW1t, nb1, nW2t, nb2, nW3t, nb3, ng, nbt,
        out_node, N);
}